// SocialPooling_3169685865097
// MI455X (gfx1250) — compile-verified
//
#include <hip/hip_runtime.h>
#include <hip/hip_bf16.h>

typedef __attribute__((ext_vector_type(2))) float v2f;
typedef __attribute__((ext_vector_type(8))) float v8f;

#define HD          64            // hidden dim
#define NPED        64            // peds per sequence
#define NSEQ        256
#define NTOT        (NSEQ * NPED) // 16384 rows
#define NCELL       64            // 8x8 grid
#define TILE_M      16            // peds per block
#define CHUNK_C     8             // cells per chunk
#define CHUNK_K     (CHUNK_C * HD)        // 512
#define POOL_PITCH  (CHUNK_K + 4)         // +4 floats: de-conflict stride-512 banks
#define EPSV        1e-5f

// ---------------------------------------------------------------------------
// Kernel 1: social pooling (LDS scatter) + fp32 WMMA GEMM + per-block stats
// 1024 blocks x 128 threads (4 waves). Block b: seq = b>>2, ped tile = b&3.
// ---------------------------------------------------------------------------
__global__ __launch_bounds__(128) void sp_pool_gemm(
    const float* __restrict__ h,        // (NTOT, 64)
    const float* __restrict__ pos,      // (NTOT, 2)
    const float* __restrict__ W,        // (4096, 64)
    const float* __restrict__ bias,     // (64)
    float* __restrict__ xout,           // (NTOT, 64) raw x
    float* __restrict__ partials)       // (1024, 2, 64) block sums / sumsq
{
    __shared__ float sh_h[NPED][HD];             // 16 KB
    __shared__ float sh_px[NPED], sh_py[NPED];
    __shared__ int   sh_cell[TILE_M][NPED];      // 4 KB, -1 = invalid
    __shared__ float sh_pool[TILE_M][POOL_PITCH];// ~33 KB

    const int tid  = threadIdx.x;
    const int wave = tid >> 5;
    const int lane = tid & 31;
    const int seq  = blockIdx.x >> 2;
    const int tile = blockIdx.x & 3;

    // ---- stage h (as float4) and positions for this sequence ----
    {
        const float4* h4 = (const float4*)h;
        float4* s4 = (float4*)&sh_h[0][0];
        for (int i = tid; i < NPED * HD / 4; i += 128)
            s4[i] = h4[(size_t)seq * (NPED * HD / 4) + i];
        if (tid < NPED) {
            sh_px[tid] = pos[(size_t)(seq * NPED + tid) * 2 + 0];
            sh_py[tid] = pos[(size_t)(seq * NPED + tid) * 2 + 1];
        }
    }
    __syncthreads();

    // ---- cell ids for all (i_local, j) pairs ----
    for (int p = tid; p < TILE_M * NPED; p += 128) {
        int il = p >> 6, j = p & 63;
        int ig = tile * TILE_M + il;                 // ped index within sequence
        float cx = sh_px[ig], cy = sh_py[ig];
        float ox = sh_px[j],  oy = sh_py[j];
        float tlx = cx - 1.0f, tly = cy + 1.0f;
        float brx = cx + 1.0f, bry = cy - 1.0f;
        bool valid = (ox < brx) & (ox > tlx) & (oy < tly) & (oy > bry) & (j != ig);
        int cell = -1;
        if (valid) {
            int gx = (int)floorf((ox - tlx) * 4.0f);   // GRID/NS = 4
            int gy = (int)floorf((tly - oy) * 4.0f);
            cell = min(max(gx + gy * 8, 0), NCELL - 1);
        }
        sh_cell[il][j] = cell;
    }

    v8f acc = {};                       // wave's 16x16 fp32 output tile
    const int n0    = wave * 16;        // this wave's output-column tile
    const int nlane = lane & 15;
    const int khalf = (lane >> 4) * 2;  // lanes 16-31 hold K+2/K+3

    for (int c = 0; c < NCELL / CHUNK_C; ++c) {
        __syncthreads();                // prior GEMM reads done
        for (int i = tid; i < TILE_M * POOL_PITCH; i += 128)
            (&sh_pool[0][0])[i] = 0.0f;
        __syncthreads();

        // ---- scatter-add: wave owns rows [4*wave, 4*wave+4) -> no atomics ----
        for (int ii = 0; ii < 4; ++ii) {
            int il = wave * 4 + ii;
            for (int j = 0; j < NPED; ++j) {
                int cc = sh_cell[il][j] - c * CHUNK_C;   // wave-uniform
                if ((unsigned)cc < CHUNK_C) {
                    int f = lane * 2;
                    float2 hv = *(const float2*)&sh_h[j][f];
                    float* pp = &sh_pool[il][cc * HD + f];
                    pp[0] += hv.x; pp[1] += hv.y;
                }
            }
        }
        __syncthreads();

        // ---- fp32 WMMA over this K-chunk (512) : 128 x (16x16x4) ----
        const float* Wc = W + (size_t)c * CHUNK_K * HD;
        const int m = lane & 15;        // A: lane -> row M
        #pragma unroll 4
        for (int kk = 0; kk < CHUNK_K / 4; ++kk) {
            v2f a = *(const v2f*)&sh_pool[m][kk * 4 + khalf];
            int r = kk * 4 + khalf;
            v2f b;
            b.x = Wc[(size_t)r * HD       + n0 + nlane];
            b.y = Wc[(size_t)(r + 1) * HD + n0 + nlane];
            acc = __builtin_amdgcn_wmma_f32_16x16x4_f32(
                false, a, false, b, (short)0, acc, false, false);
        }
    }

    // ---- epilogue: bias, store raw x, deterministic per-block stats ----
    const int n  = n0 + nlane;
    const float bn = bias[n];
    const int mbase = (lane >> 4) * 8;  // C layout: lanes 16-31 hold M+8
    float csum = 0.0f, csq = 0.0f;
    #pragma unroll
    for (int v = 0; v < 8; ++v) {
        float x = acc[v] + bn;
        int row = seq * NPED + tile * TILE_M + (v + mbase);
        xout[(size_t)row * HD + n] = x;
        csum += x; csq += x * x;
    }
    csum += __shfl_xor(csum, 16, 32);
    csq  += __shfl_xor(csq,  16, 32);
    if (lane < 16) {
        partials[((size_t)blockIdx.x * 2 + 0) * HD + n] = csum;
        partials[((size_t)blockIdx.x * 2 + 1) * HD + n] = csq;
    }
}

// ---------------------------------------------------------------------------
// Kernel 2: reduce 1024 block partials -> mu[f], rstd[f]
// ---------------------------------------------------------------------------
__global__ void sp_stats(const float* __restrict__ partials,
                         float* __restrict__ stats)   // [0:64)=mu, [64:128)=rstd
{
    int f = threadIdx.x;
    if (f < HD) {
        float s = 0.0f, q = 0.0f;
        for (int b = 0; b < 1024; ++b) {
            s += partials[((size_t)b * 2 + 0) * HD + f];
            q += partials[((size_t)b * 2 + 1) * HD + f];
        }
        const float invN = 1.0f / (float)NTOT;
        float mu  = s * invN;
        float var = fmaxf(q * invN - mu * mu, 0.0f);
        stats[f]      = mu;
        stats[HD + f] = 1.0f / sqrtf(var + EPSV);
    }
}

// ---------------------------------------------------------------------------
// Kernel 3: in-place normalize + affine + ReLU
// ---------------------------------------------------------------------------
__global__ void sp_norm_relu(float* __restrict__ x,
                             const float* __restrict__ stats,
                             const float* __restrict__ gamma,
                             const float* __restrict__ beta)
{
    int idx = blockIdx.x * blockDim.x + threadIdx.x;
    int f = idx & (HD - 1);
    float v = x[idx];
    float y = gamma[f] * (v - stats[f]) * stats[HD + f] + beta[f];
    x[idx] = fmaxf(y, 0.0f);
}

// ---------------------------------------------------------------------------
extern "C" void kernel_launch(void* const* d_in, const int* in_sizes, int n_in,
                              void* d_out, int out_size, void* d_ws, size_t ws_size,
                              hipStream_t stream) {
    const float* h     = (const float*)d_in[0];   // h_states (1, N, 64)
    // d_in[1] = seq_start_end (unused: starts are s*64 by construction)
    const float* pos   = (const float*)d_in[2];   // end_pos (N, 2)
    const float* W     = (const float*)d_in[3];   // (4096, 64)
    const float* bias  = (const float*)d_in[4];   // (64)
    const float* gamma = (const float*)d_in[5];   // (64)
    const float* beta  = (const float*)d_in[6];   // (64)

    float* xout     = (float*)d_out;              // (16384, 64)
    float* partials = (float*)d_ws;               // 1024*2*64 floats = 512 KB
    float* stats    = partials + 1024 * 2 * HD;   // 128 floats

    sp_pool_gemm<<<NTOT / TILE_M, 128, 0, stream>>>(h, pos, W, bias, xout, partials);
    sp_stats<<<1, 64, 0, stream>>>(partials, stats);
    sp_norm_relu<<<(NTOT * HD) / 256, 256, 0, stream>>>(xout, stats, gamma, beta);
}